// StochasticNonLinearIntensityTransformation_66752381714444
// MI455X (gfx1250) — compile-verified
//
#include <hip/hip_runtime.h>
#include <math.h>

// ---------------------------------------------------------------------------
// StochasticNonLinearIntensityTransformation for MI455X (gfx1250, wave32)
//
// Bandwidth-bound plan (~768MB traffic => ~33us floor @ 23.3 TB/s):
//   k0: init per-batch min/max atomic slots in d_ws
//   k1: stream image (128b loads, RT hint -> keep in 192MB L2), interp via
//       LDS LUT, wave-shuffle + LDS block reduce, atomic per-batch min/max
//   k2: re-stream image (NT loads, last use), recompute interp, normalize,
//       NT 128b stores (output never re-read; don't evict pass-1 lines)
// The 512-entry Bezier LUT is a (1x4)x(4x512) GEMM -> built per block in LDS
// with V_WMMA_F32_16X16X4_F32 (32 tiles of 16 columns, spread over 8 waves).
// ---------------------------------------------------------------------------

typedef float v2f __attribute__((ext_vector_type(2)));
typedef float v4f __attribute__((ext_vector_type(4)));
typedef float v8f __attribute__((ext_vector_type(8)));

#define LUT               512
#define BATCH             64
#define HW                (1024 * 1024)
#define THREADS           256
#define WAVES             (THREADS / 32)
#define BLOCKS_PER_BATCH  256
#define ELEMS_PER_BLOCK   (HW / BLOCKS_PER_BATCH)          // 4096
#define VEC_PER_THREAD    (ELEMS_PER_BLOCK / (4 * THREADS)) // 4 x 128b / thread

// ---- order-preserving float<->uint mapping for atomic min/max -------------
__device__ __forceinline__ unsigned enc_f32(float f) {
    unsigned u = __float_as_uint(f);
    return (u & 0x80000000u) ? ~u : (u | 0x80000000u);
}
__device__ __forceinline__ float dec_f32(unsigned o) {
    unsigned u = (o & 0x80000000u) ? (o & 0x7FFFFFFFu) : ~o;
    return __uint_as_float(u);
}

// ---- build 512-entry clipped Bezier LUT in LDS via WMMA f32 16x16x4 -------
// curve[j] = clip( sum_k cpY[k] * bern_k(t_j), -1, 1 ),  t_j = -1 + j*2/511
// A (16x4, all rows = cpY): lane<16 holds K=0,1 in v[0],v[1]; lane>=16 K=2,3.
// B (4x16): lane 0..15 = column j of tile; same K split as A.
// D row 0 lives in VGPR0 of lanes 0..15 -> the 16 curve values of the tile.
__device__ __forceinline__ void build_curve_lds(float* curve,
                                                const float* __restrict__ cp) {
    const float y0 = cp[1], y1 = cp[3], y2 = cp[5], y3 = cp[7]; // cp[k][1]
    const int lane = threadIdx.x & 31;
    const int wave = threadIdx.x >> 5;       // 0..7
    const bool hi  = lane >= 16;

    v2f a;
    a.x = hi ? y2 : y0;
    a.y = hi ? y3 : y1;

    const float h = 2.0f / 511.0f;
#pragma unroll
    for (int tile = 0; tile < 4; ++tile) {
        const int tt = wave + tile * 8;              // tiles 0..31 over 8 waves
        const int j  = tt * 16 + (lane & 15);
        const float t = -1.0f + h * (float)j;
        const float s = 1.0f - t;
        // Selects (not branches) feed the WMMA so EXEC stays all-ones
        const float b_lo0 = s * s * s;
        const float b_lo1 = 3.0f * t * s * s;
        const float b_hi0 = 3.0f * t * t * s;
        const float b_hi1 = t * t * t;
        v2f b;
        b.x = hi ? b_hi0 : b_lo0;
        b.y = hi ? b_hi1 : b_lo1;

        v8f c = {};
        v8f d = __builtin_amdgcn_wmma_f32_16x16x4_f32(
            /*neg_a=*/false, a, /*neg_b=*/false, b,
            /*c_mod=*/(short)0, c, /*reuse_a=*/false, /*reuse_b=*/false);

        if (lane < 16) {
            float v = d[0];                           // D[0][lane] = curve[j]
            v = fminf(1.0f, fmaxf(-1.0f, v));
            curve[tt * 16 + lane] = v;
        }
    }
    __syncthreads();
}

// ---- analytic searchsorted on the uniform grid + LDS LUT lerp -------------
__device__ __forceinline__ float interp_one(float x,
                                            const float* __restrict__ curve) {
    const float inv_h = 511.0f * 0.5f;               // 1/h
    const float h     = 2.0f / 511.0f;
    int idx = (int)ceilf((x + 1.0f) * inv_h);        // searchsorted 'left'
    idx = min(511, max(1, idx));
    const float x0 = -1.0f + h * (float)(idx - 1);
    const float yA = curve[idx - 1];
    const float yB = curve[idx];
    return fmaf((yB - yA) * inv_h, x - x0, yA);
}

// ---- k0: reset per-batch atomic slots every launch (graph-replay safe) ----
__global__ void init_ws_kernel(unsigned* __restrict__ ws) {
    const int t = threadIdx.x;
    if (t < BATCH)            ws[t] = 0xFFFFFFFFu;   // min slots
    else if (t < 2 * BATCH)   ws[t] = 0x00000000u;   // max slots
}

// ---- k1: interp + per-batch min/max reduction -----------------------------
__global__ void __launch_bounds__(THREADS)
minmax_kernel(const float* __restrict__ img, const float* __restrict__ cp,
              unsigned* __restrict__ wsMin, unsigned* __restrict__ wsMax) {
    __shared__ float curve[LUT];
    __shared__ float wmin[WAVES];
    __shared__ float wmax[WAVES];

    build_curve_lds(curve, cp);

    const int batch = blockIdx.y;
    const size_t base = (size_t)batch * HW + (size_t)blockIdx.x * ELEMS_PER_BLOCK;
    const v4f* __restrict__ pin = (const v4f*)(img + base);

    float mn =  INFINITY;
    float mx = -INFINITY;
#pragma unroll
    for (int i = 0; i < VEC_PER_THREAD; ++i) {
        const v4f v = pin[threadIdx.x + i * THREADS]; // global_load_b128 RT
        const float r0 = interp_one(v.x, curve);
        const float r1 = interp_one(v.y, curve);
        const float r2 = interp_one(v.z, curve);
        const float r3 = interp_one(v.w, curve);
        mn = fminf(mn, fminf(fminf(r0, r1), fminf(r2, r3)));
        mx = fmaxf(mx, fmaxf(fmaxf(r0, r1), fmaxf(r2, r3)));
    }

    // wave32 shuffle reduction (5 steps), then 8 partials through LDS
    const int lane = threadIdx.x & 31;
    const int wave = threadIdx.x >> 5;
#pragma unroll
    for (int m = 16; m > 0; m >>= 1) {
        mn = fminf(mn, __shfl_xor(mn, m, 32));
        mx = fmaxf(mx, __shfl_xor(mx, m, 32));
    }
    if (lane == 0) { wmin[wave] = mn; wmax[wave] = mx; }
    __syncthreads();
    if (threadIdx.x == 0) {
        float bmn = wmin[0], bmx = wmax[0];
#pragma unroll
        for (int w = 1; w < WAVES; ++w) {
            bmn = fminf(bmn, wmin[w]);
            bmx = fmaxf(bmx, wmax[w]);
        }
        atomicMin(&wsMin[batch], enc_f32(bmn));
        atomicMax(&wsMax[batch], enc_f32(bmx));
    }
}

// ---- k2: recompute interp, normalize, store -------------------------------
__global__ void __launch_bounds__(THREADS)
apply_kernel(const float* __restrict__ img, const float* __restrict__ cp,
             const unsigned* __restrict__ wsMin,
             const unsigned* __restrict__ wsMax,
             float* __restrict__ out) {
    __shared__ float curve[LUT];
    build_curve_lds(curve, cp);

    const int batch = blockIdx.y;
    const float mn = dec_f32(wsMin[batch]);
    const float mx = dec_f32(wsMax[batch]);
    const float sc  = 2.0f / (mx - mn);
    const float off = fmaf(-mn, sc, -1.0f);          // out = interp*sc + off

    const size_t base = (size_t)batch * HW + (size_t)blockIdx.x * ELEMS_PER_BLOCK;
    const v4f* __restrict__ pin  = (const v4f*)(img + base);
    v4f* __restrict__       pout = (v4f*)(out + base);

#pragma unroll
    for (int i = 0; i < VEC_PER_THREAD; ++i) {
        const int k = threadIdx.x + i * THREADS;
        // last-use read: NT hint, may still hit pass-1 lines in L2
        const v4f v = __builtin_nontemporal_load(&pin[k]);
        v4f r;
        r.x = fmaf(interp_one(v.x, curve), sc, off);
        r.y = fmaf(interp_one(v.y, curve), sc, off);
        r.z = fmaf(interp_one(v.z, curve), sc, off);
        r.w = fmaf(interp_one(v.w, curve), sc, off);
        // streaming write: never re-read, keep it out of L2's way
        __builtin_nontemporal_store(r, &pout[k]);
    }
}

extern "C" void kernel_launch(void* const* d_in, const int* in_sizes, int n_in,
                              void* d_out, int out_size, void* d_ws, size_t ws_size,
                              hipStream_t stream) {
    (void)in_sizes; (void)n_in; (void)out_size; (void)ws_size;
    const float* img = (const float*)d_in[0];        // (64,1,1024,1024) f32
    const float* cp  = (const float*)d_in[1];        // (4,2) f32
    float* out       = (float*)d_out;
    unsigned* ws     = (unsigned*)d_ws;
    unsigned* wsMin  = ws;
    unsigned* wsMax  = ws + BATCH;

    hipLaunchKernelGGL(init_ws_kernel, dim3(1), dim3(2 * BATCH), 0, stream, ws);

    const dim3 grid(BLOCKS_PER_BATCH, BATCH);
    const dim3 block(THREADS);
    hipLaunchKernelGGL(minmax_kernel, grid, block, 0, stream, img, cp, wsMin, wsMax);
    hipLaunchKernelGGL(apply_kernel,  grid, block, 0, stream, img, cp, wsMin, wsMax, out);
}